// GatedMoE_80032420593967
// MI455X (gfx1250) — compile-verified
//
#include <hip/hip_runtime.h>
#include <hip/hip_bf16.h>

typedef __attribute__((ext_vector_type(2))) float v2f;
typedef __attribute__((ext_vector_type(8))) float v8f;

#define NTOK 2048   // B*T = 4*512
#define DDIM 128
#define EEXP 256
#define XS_LD 132   // padded LDS row stride (floats): 132*m mod 64 banks = 4m -> conflict-free A reads
#define LG_LD 260   // padded logits row stride

// ---------------------------------------------------------------------------
// Kernel 1: router. One WG (8 waves) per 16-token tile.
// logits[16,256] = x[16,128] @ gate_w[128,256] + gate_b via V_WMMA_F32_16X16X4_F32
// then per-token softmax + top-2 + renormalized weights.
// ---------------------------------------------------------------------------
__global__ __launch_bounds__(256) void moe_router_kernel(
    const float* __restrict__ x, const float* __restrict__ gate_w,
    const float* __restrict__ gate_b,
    int* __restrict__ top_e, float* __restrict__ top_w)
{
  __shared__ float xs[16 * XS_LD];
  __shared__ float lg[16 * LG_LD];
  const int tid  = threadIdx.x;
  const int tok0 = blockIdx.x * 16;

  // stage x tile with b128 loads (16 rows x 32 float4)
  for (int i = tid; i < 16 * (DDIM / 4); i += 256) {
    const int r = i >> 5, c4 = i & 31;
    const float4 v = reinterpret_cast<const float4*>(x + (size_t)(tok0 + r) * DDIM)[c4];
    reinterpret_cast<float4*>(&xs[r * XS_LD])[c4] = v;
  }
  __syncthreads();

  const int wave = tid >> 5, lane = tid & 31;
  const int n = lane & 15, hi = lane >> 4, mrow = lane & 15;

  for (int ct = wave; ct < EEXP / 16; ct += 8) {
    const int col0 = ct * 16;
    v8f acc = {};
#pragma unroll
    for (int k0 = 0; k0 < DDIM; k0 += 4) {
      v2f a, b;
      a.x = xs[mrow * XS_LD + k0 + 2 * hi];
      a.y = xs[mrow * XS_LD + k0 + 2 * hi + 1];
      b.x = gate_w[(k0 + 2 * hi) * EEXP + col0 + n];
      b.y = gate_w[(k0 + 2 * hi + 1) * EEXP + col0 + n];
      acc = __builtin_amdgcn_wmma_f32_16x16x4_f32(false, a, false, b,
                                                  (short)0, acc, false, false);
    }
    const float bias = gate_b[col0 + n];
#pragma unroll
    for (int j = 0; j < 8; ++j)
      lg[(j + 8 * hi) * LG_LD + col0 + n] = acc[j] + bias;
  }
  __syncthreads();

  if (tid < 16) {
    const int t = tok0 + tid;
    const float* L = &lg[tid * LG_LD];
    float m = -3.4e38f;
    for (int e = 0; e < EEXP; ++e) m = fmaxf(m, L[e]);
    float Z = 0.f, v1 = -1.f, v2 = -1.f;
    int i1 = 0, i2 = 0;
    for (int e = 0; e < EEXP; ++e) {
      float p = __expf(L[e] - m);
      Z += p;
      if (p > v1)      { v2 = v1; i2 = i1; v1 = p; i1 = e; }
      else if (p > v2) { v2 = p;  i2 = e; }
    }
    v1 /= Z; v2 /= Z;                       // softmax top-2 values
    const float s = v1 + v2 + 1e-6f;        // renormalize (matches reference)
    top_e[t * 2 + 0] = i1;  top_e[t * 2 + 1] = i2;
    top_w[t * 2 + 0] = v1 / s;  top_w[t * 2 + 1] = v2 / s;
  }
}

// ---------------------------------------------------------------------------
// Kernel 2: expert GEMM. One WG per expert; gather this expert's tokens,
// GEMM 16-token tiles against expert_w[e] (128x128) with f32 WMMA,
// store unscaled y into ybuf[(token*2 + slot)*128 + col].
// Row results are bitwise independent of list order, so the LDS-atomic
// append order does not affect the output (deterministic).
// ---------------------------------------------------------------------------
__global__ __launch_bounds__(256) void moe_expert_kernel(
    const float* __restrict__ x, const float* __restrict__ expert_w,
    const int* __restrict__ top_e, float* __restrict__ ybuf)
{
  __shared__ int   toklist[NTOK];   // token*2 + slot
  __shared__ float xs[16 * XS_LD];
  __shared__ int   cnt;

  const int e = blockIdx.x;
  const int tid = threadIdx.x;
  if (tid == 0) cnt = 0;
  __syncthreads();

  for (int t = tid; t < NTOK; t += 256) {
    if (top_e[t * 2 + 0] == e) { int p = atomicAdd(&cnt, 1); toklist[p] = t * 2; }
    if (top_e[t * 2 + 1] == e) { int p = atomicAdd(&cnt, 1); toklist[p] = t * 2 + 1; }
  }
  __syncthreads();
  const int count = cnt;
  const float* __restrict__ W = expert_w + (size_t)e * DDIM * DDIM;

  const int wave = tid >> 5, lane = tid & 31;
  const int n = lane & 15, hi = lane >> 4, mrow = lane & 15;
  const int col0 = wave * 16;   // 8 waves cover 128 output cols

  for (int base = 0; base < count; base += 16) {
    __syncthreads();
    // stage 16 gathered rows with b128 loads
    for (int i = tid; i < 16 * (DDIM / 4); i += 256) {
      const int r = i >> 5, c4 = i & 31;
      const int idx = base + r;
      float4 v = make_float4(0.f, 0.f, 0.f, 0.f);
      if (idx < count)
        v = reinterpret_cast<const float4*>(x + (size_t)(toklist[idx] >> 1) * DDIM)[c4];
      reinterpret_cast<float4*>(&xs[r * XS_LD])[c4] = v;
    }
    __syncthreads();

    v8f acc = {};
#pragma unroll
    for (int k0 = 0; k0 < DDIM; k0 += 4) {
      v2f a, b;
      a.x = xs[mrow * XS_LD + k0 + 2 * hi];
      a.y = xs[mrow * XS_LD + k0 + 2 * hi + 1];
      b.x = W[(k0 + 2 * hi) * DDIM + col0 + n];
      b.y = W[(k0 + 2 * hi + 1) * DDIM + col0 + n];
      acc = __builtin_amdgcn_wmma_f32_16x16x4_f32(false, a, false, b,
                                                  (short)0, acc, false, false);
    }
#pragma unroll
    for (int j = 0; j < 8; ++j) {
      const int idx = base + j + 8 * hi;
      if (idx < count)
        ybuf[(size_t)toklist[idx] * DDIM + col0 + n] = acc[j];
    }
  }
}

// ---------------------------------------------------------------------------
// Kernel 3: SwiGLU. pre = w1n*y1 + w2n*y2 (deterministic, no atomics),
// out = swish(pre@W1 + b1) * (pre@W2 + b2), both GEMMs via f32 WMMA.
// ---------------------------------------------------------------------------
__global__ __launch_bounds__(256) void moe_swiglu_kernel(
    const float* __restrict__ ybuf, const float* __restrict__ top_w,
    const float* __restrict__ w1, const float* __restrict__ b1,
    const float* __restrict__ w2, const float* __restrict__ b2,
    float* __restrict__ out)
{
  __shared__ float as[16 * XS_LD];
  const int tid  = threadIdx.x;
  const int tok0 = blockIdx.x * 16;

  for (int i = tid; i < 16 * (DDIM / 4); i += 256) {
    const int r = i >> 5, c4 = i & 31;
    const int t = tok0 + r;
    const float wa = top_w[t * 2 + 0], wb = top_w[t * 2 + 1];
    const float4 y1 = reinterpret_cast<const float4*>(ybuf + (size_t)(t * 2) * DDIM)[c4];
    const float4 y2 = reinterpret_cast<const float4*>(ybuf + (size_t)(t * 2 + 1) * DDIM)[c4];
    float4 v;
    v.x = wa * y1.x + wb * y2.x;  v.y = wa * y1.y + wb * y2.y;
    v.z = wa * y1.z + wb * y2.z;  v.w = wa * y1.w + wb * y2.w;
    reinterpret_cast<float4*>(&as[r * XS_LD])[c4] = v;
  }
  __syncthreads();

  const int wave = tid >> 5, lane = tid & 31;
  const int n = lane & 15, hi = lane >> 4, mrow = lane & 15;
  const int col0 = wave * 16;

  v8f acc1 = {}, acc2 = {};
#pragma unroll
  for (int k0 = 0; k0 < DDIM; k0 += 4) {
    v2f a, bA, bB;
    a.x = as[mrow * XS_LD + k0 + 2 * hi];
    a.y = as[mrow * XS_LD + k0 + 2 * hi + 1];
    bA.x = w1[(k0 + 2 * hi) * DDIM + col0 + n];
    bA.y = w1[(k0 + 2 * hi + 1) * DDIM + col0 + n];
    acc1 = __builtin_amdgcn_wmma_f32_16x16x4_f32(false, a, false, bA,
                                                 (short)0, acc1, false, false);
    bB.x = w2[(k0 + 2 * hi) * DDIM + col0 + n];
    bB.y = w2[(k0 + 2 * hi + 1) * DDIM + col0 + n];
    acc2 = __builtin_amdgcn_wmma_f32_16x16x4_f32(false, a, false, bB,
                                                 (short)0, acc2, false, false);
  }
  const float bb1 = b1[col0 + n], bb2 = b2[col0 + n];
#pragma unroll
  for (int j = 0; j < 8; ++j) {
    const int t = tok0 + j + 8 * hi;
    const float g = acc1[j] + bb1;
    const float h = acc2[j] + bb2;
    const float sig = 1.f / (1.f + __expf(-g));
    out[(size_t)t * DDIM + col0 + n] = g * sig * h;
  }
}

// ---------------------------------------------------------------------------
// Kernel 4: consensus = exp(-mean_d( w1*(y1-out)^2 + w2*(y2-out)^2 ))
// ---------------------------------------------------------------------------
__global__ __launch_bounds__(128) void moe_consensus_kernel(
    const float* __restrict__ ybuf, const float* __restrict__ top_w,
    const float* __restrict__ out, float* __restrict__ consensus)
{
  __shared__ float red[DDIM];
  const int t = blockIdx.x;
  const int d = threadIdx.x;
  const float o  = out[(size_t)t * DDIM + d];
  const float d1 = ybuf[(size_t)(t * 2) * DDIM + d] - o;
  const float d2 = ybuf[(size_t)(t * 2 + 1) * DDIM + d] - o;
  red[d] = top_w[t * 2 + 0] * d1 * d1 + top_w[t * 2 + 1] * d2 * d2;
  __syncthreads();
  for (int s = 64; s > 0; s >>= 1) {
    if (d < s) red[d] += red[d + s];
    __syncthreads();
  }
  if (d == 0) consensus[t] = __expf(-red[0] * (1.f / (float)DDIM));
}

// ---------------------------------------------------------------------------
extern "C" void kernel_launch(void* const* d_in, const int* in_sizes, int n_in,
                              void* d_out, int out_size, void* d_ws, size_t ws_size,
                              hipStream_t stream) {
  (void)in_sizes; (void)n_in; (void)out_size; (void)ws_size;
  const float* x        = (const float*)d_in[0];
  const float* gate_w   = (const float*)d_in[1];
  const float* gate_b   = (const float*)d_in[2];
  const float* expert_w = (const float*)d_in[3];
  const float* w1       = (const float*)d_in[4];
  const float* b1       = (const float*)d_in[5];
  const float* w2       = (const float*)d_in[6];
  const float* b2       = (const float*)d_in[7];

  // Workspace layout (all rewritten every call; no cross-call state):
  int*   top_e = (int*)d_ws;                         // 2048*2 ints
  float* top_w = (float*)d_ws + NTOK * 2;            // 2048*2 floats
  float* ybuf  = (float*)d_ws + NTOK * 4;            // 2048*2*128 floats (~2.1 MB)

  float* avg_out  = (float*)d_out;                   // [B,T,D] = 2048*128
  float* cons_out = (float*)d_out + NTOK * DDIM;     // [B,T]   = 2048

  moe_router_kernel<<<NTOK / 16, 256, 0, stream>>>(x, gate_w, gate_b, top_e, top_w);
  moe_expert_kernel<<<EEXP, 256, 0, stream>>>(x, expert_w, top_e, ybuf);
  moe_swiglu_kernel<<<NTOK / 16, 256, 0, stream>>>(ybuf, top_w, w1, b1, w2, b2, avg_out);
  moe_consensus_kernel<<<NTOK, 128, 0, stream>>>(ybuf, top_w, avg_out, cons_out);
}